// HeadDynamicK_79053168050560
// MI455X (gfx1250) — compile-verified
//
#include <hip/hip_runtime.h>
#include <hip/hip_bf16.h>

// ---------------------------------------------------------------------------
// SparseRCNN dynamic-conv head for MI455X (gfx1250), fp32 end-to-end.
// All GEMMs on V_WMMA_F32_16X16X4_F32; waves hold multiple C tiles per B
// fragment (5x / 2-4x / 2x) for WMMA density; all LDS staging is b128
// vectorized; out-layer K-chunks are double-buffered with register prefetch.
//   H=256 hidden, D=64 dynamic dim, R=49 pooled pixels, N=2000 instances.
// ---------------------------------------------------------------------------

constexpr int Hh = 256;   // hidden dim
constexpr int Dd = 64;    // dynamic dim
constexpr int Rr = 49;    // pooler res^2
constexpr int Nn = 2000;  // instances
constexpr int PW = 2 * Hh * Dd;   // 32768 params per instance
constexpr int KO = Rr * Hh;       // 12544, out-layer K
constexpr int NC = KO / 256;      // 49 K-chunks in out-layer

typedef __attribute__((ext_vector_type(2))) float v2f;
typedef __attribute__((ext_vector_type(8))) float v8f;

__device__ __forceinline__ v8f wmma_f32(v2f a, v2f b, v8f c) {
  return __builtin_amdgcn_wmma_f32_16x16x4_f32(
      false, a, false, b, (short)0, c, false, false);
}

// ---------------------------------------------------------------------------
// Kernel 1: params = pro(2000x256) @ W_dyn(256x32768) + b_dyn
// grid = (25 M-tiles of 80, 256 N-tiles of 128), block = 256 (8 waves).
// Each wave: 5 M-tiles x 1 N-tile -> 5 wmma per B fragment fetch.
// ---------------------------------------------------------------------------
__global__ __launch_bounds__(256) void params_gemm_k(
    const float* __restrict__ pro, const float* __restrict__ Wdyn,
    const float* __restrict__ bdyn, float* __restrict__ params) {
  const int tid  = threadIdx.x;
  const int wave = tid >> 5;
  const int lane = tid & 31;

  __shared__ float As[80][260];              // stride 260: 16B-aligned rows,
                                             // bank stride 4 -> conflict-free
  const int m0 = blockIdx.x * 80;
  for (int i4 = tid; i4 < 80 * 64; i4 += 256) {
    const int r = i4 >> 6, c4 = (i4 & 63) * 4;
    *(float4*)&As[r][c4] = *(const float4*)&pro[(size_t)(m0 + r) * Hh + c4];
  }
  __syncthreads();

  const int ncol  = blockIdx.y * 128 + wave * 16 + (lane & 15);
  const int mrow  = lane & 15;
  const int khalf = (lane >> 4) * 2;

  v8f acc[5];
#pragma unroll
  for (int i = 0; i < 5; ++i) acc[i] = (v8f){};

  const float* Bp = Wdyn + ncol;
  for (int k = 0; k < Hh; k += 4) {
    const int kk = k + khalf;
    v2f b = { Bp[(size_t)kk * PW], Bp[(size_t)(kk + 1) * PW] };
#pragma unroll
    for (int i = 0; i < 5; ++i) {
      v2f a = { As[i * 16 + mrow][kk], As[i * 16 + mrow][kk + 1] };
      acc[i] = wmma_f32(a, b, acc[i]);
    }
  }

  const float bias = bdyn[ncol];
#pragma unroll
  for (int i = 0; i < 5; ++i) {
    const int mbase = m0 + i * 16 + (lane >> 4) * 8;
#pragma unroll
    for (int g = 0; g < 8; ++g)
      params[(size_t)(mbase + g) * PW + ncol] = acc[i][g] + bias;
  }
}

// ---------------------------------------------------------------------------
// Kernel 2: per-instance fused  f1 = feats@p1 ; LN(64)+ReLU ;
//                               f2 = f1@p2    ; LN(256)+ReLU -> workspace
// grid = 2000, block = 256 (8 waves). R padded 49 -> 64 with zero rows that
// stay exactly zero through both WMMAs.
// ---------------------------------------------------------------------------
__global__ __launch_bounds__(256) void dynconv_k(
    const float* __restrict__ roi, const float* __restrict__ params,
    const float* __restrict__ g1, const float* __restrict__ b1,
    const float* __restrict__ g2, const float* __restrict__ b2,
    float* __restrict__ f2out) {
  const int n    = blockIdx.x;
  const int tid  = threadIdx.x;
  const int wave = tid >> 5;
  const int lane = tid & 31;

  __shared__ float Fs[64][260];              // feats, later reused as f2
  __shared__ float F1[64][68];               // bmm1 result

  for (int i4 = tid; i4 < 64 * 64; i4 += 256) {
    const int r = i4 >> 6, c4 = (i4 & 63) * 4;
    float4 v = make_float4(0.f, 0.f, 0.f, 0.f);
    if (r < Rr)
      v = *(const float4*)&roi[(size_t)r * (Nn * Hh) + (size_t)n * Hh + c4];
    *(float4*)&Fs[r][c4] = v;
  }
  __syncthreads();

  const int khalf = (lane >> 4) * 2;
  const float* p1 = params + (size_t)n * PW;            // (256 x 64)
  const float* p2 = p1 + Hh * Dd;                       // (64 x 256)

  // ---- bmm1: F1(64x64) = Fs(64x256) @ p1(256x64) ----
  // wave w: N-tile = w&3, M-tiles = {(w>>2)*2, (w>>2)*2+1}; B reused 2x.
  {
    const int d     = (wave & 3) * 16 + (lane & 15);
    const int mrow0 = ((wave >> 2) * 2) * 16 + (lane & 15);
    v8f acc0 = {}, acc1 = {};
    for (int k = 0; k < Hh; k += 4) {
      const int kk = k + khalf;
      v2f b  = { p1[(size_t)kk * Dd + d], p1[(size_t)(kk + 1) * Dd + d] };
      v2f a0 = { Fs[mrow0][kk],      Fs[mrow0][kk + 1] };
      v2f a1 = { Fs[mrow0 + 16][kk], Fs[mrow0 + 16][kk + 1] };
      acc0 = wmma_f32(a0, b, acc0);
      acc1 = wmma_f32(a1, b, acc1);
    }
    const int mbase = ((wave >> 2) * 2) * 16 + (lane >> 4) * 8;
#pragma unroll
    for (int g = 0; g < 8; ++g) {
      F1[mbase + g][d]      = acc0[g];
      F1[mbase + 16 + g][d] = acc1[g];
    }
  }
  __syncthreads();

  // ---- LayerNorm over D=64 + ReLU on the 49 valid rows ----
  if (tid < Rr) {
    float m = 0.f;
    for (int d2 = 0; d2 < Dd; ++d2) m += F1[tid][d2];
    m *= (1.0f / Dd);
    float v = 0.f;
    for (int d2 = 0; d2 < Dd; ++d2) { const float x = F1[tid][d2] - m; v += x * x; }
    v *= (1.0f / Dd);
    const float rs = rsqrtf(v + 1e-5f);
    for (int d2 = 0; d2 < Dd; ++d2) {
      const float x = (F1[tid][d2] - m) * rs * g1[d2] + b1[d2];
      F1[tid][d2] = fmaxf(x, 0.0f);
    }
  }
  __syncthreads();

  // ---- bmm2: Fs(64x256) = F1(64x64) @ p2(64x256) ----
  // wave w: N-tiles {2w, 2w+1}, all 4 M-tiles; each B fragment reused 4x.
  {
    const int h0 = (wave * 2) * 16 + (lane & 15);
    const int h1 = h0 + 16;
    v8f acc[2][4];
#pragma unroll
    for (int j = 0; j < 2; ++j)
#pragma unroll
      for (int mt = 0; mt < 4; ++mt) acc[j][mt] = (v8f){};

    for (int k = 0; k < Dd; k += 4) {
      const int kk = k + khalf;
      v2f a[4];
#pragma unroll
      for (int mt = 0; mt < 4; ++mt) {
        const int mrow = mt * 16 + (lane & 15);
        a[mt] = (v2f){ F1[mrow][kk], F1[mrow][kk + 1] };
      }
      v2f b0  = { p2[(size_t)kk * Hh + h0], p2[(size_t)(kk + 1) * Hh + h0] };
      v2f b1v = { p2[(size_t)kk * Hh + h1], p2[(size_t)(kk + 1) * Hh + h1] };
#pragma unroll
      for (int mt = 0; mt < 4; ++mt) {
        acc[0][mt] = wmma_f32(a[mt], b0,  acc[0][mt]);
        acc[1][mt] = wmma_f32(a[mt], b1v, acc[1][mt]);
      }
    }
#pragma unroll
    for (int mt = 0; mt < 4; ++mt) {
      const int mbase = mt * 16 + (lane >> 4) * 8;
#pragma unroll
      for (int g = 0; g < 8; ++g) {
        Fs[mbase + g][h0] = acc[0][mt][g];
        Fs[mbase + g][h1] = acc[1][mt][g];
      }
    }
  }
  __syncthreads();

  // ---- LayerNorm over H=256 + ReLU ----
  if (tid < Rr) {
    float m = 0.f;
    for (int h = 0; h < Hh; ++h) m += Fs[tid][h];
    m *= (1.0f / Hh);
    float v = 0.f;
    for (int h = 0; h < Hh; ++h) { const float x = Fs[tid][h] - m; v += x * x; }
    v *= (1.0f / Hh);
    const float rs = rsqrtf(v + 1e-5f);
    for (int h = 0; h < Hh; ++h) {
      const float x = (Fs[tid][h] - m) * rs * g2[h] + b2[h];
      Fs[tid][h] = fmaxf(x, 0.0f);
    }
  }
  __syncthreads();

  for (int i4 = tid; i4 < (Rr * Hh) / 4; i4 += 256) {
    const int r = i4 >> 6, c4 = (i4 & 63) * 4;
    *(float4*)&f2out[(size_t)n * KO + r * Hh + c4] = *(const float4*)&Fs[r][c4];
  }
}

// ---------------------------------------------------------------------------
// Kernel 3: out = ReLU(LN( f2(2000x12544) @ W_out(12544x256) + b_out ))
// grid = 63 (M-tiles of 32, clamped tail), block = 512 (16 waves = all 256
// columns, so LN(256) stays intra-workgroup). Each wave: 2 M-tiles per B
// fragment. K streamed through double-buffered 32x256 LDS chunks with
// register prefetch: next chunk's b128 global loads issue before the wmma
// loop, land in the other buffer after it; one barrier per chunk.
// ---------------------------------------------------------------------------
__global__ __launch_bounds__(512) void outlayer_k(
    const float* __restrict__ f2, const float* __restrict__ Wout,
    const float* __restrict__ bout, const float* __restrict__ g3,
    const float* __restrict__ b3, float* __restrict__ out) {
  const int tid  = threadIdx.x;
  const int wave = tid >> 5;
  const int lane = tid & 31;

  __shared__ float As[2][32][260];   // double-buffered A chunk
  __shared__ float Cs[32][260];      // C staging + LN stats in the pad

  const int m0    = blockIdx.x * 32;
  const int h     = wave * 16 + (lane & 15);
  const int mrow  = lane & 15;
  const int khalf = (lane >> 4) * 2;

  // this thread's 4 float4 slots of each 32x256 chunk
  int rr[4], cc[4];
  const float* gsrc[4];
#pragma unroll
  for (int j = 0; j < 4; ++j) {
    const int idx = j * 2048 + tid * 4;
    rr[j] = idx >> 8;
    cc[j] = idx & 255;
    const int src = (m0 + rr[j] < Nn) ? (m0 + rr[j]) : (Nn - 1);   // tail clamp
    gsrc[j] = f2 + (size_t)src * KO + cc[j];
  }

  float4 rbuf[4];
#pragma unroll
  for (int j = 0; j < 4; ++j) rbuf[j] = *(const float4*)(gsrc[j]);
#pragma unroll
  for (int j = 0; j < 4; ++j) *(float4*)&As[0][rr[j]][cc[j]] = rbuf[j];
  __syncthreads();

  v8f acc0 = {}, acc1 = {};
  for (int c = 0; c < NC; ++c) {
    if (c + 1 < NC) {                 // prefetch next chunk into registers
#pragma unroll
      for (int j = 0; j < 4; ++j)
        rbuf[j] = *(const float4*)(gsrc[j] + (size_t)(c + 1) * 256);
    }
    const int buf = c & 1;
    const int kc  = c * 256;
    for (int k = 0; k < 256; k += 4) {
      const int kk = k + khalf;
      v2f b  = { Wout[(size_t)(kc + kk) * Hh + h],
                 Wout[(size_t)(kc + kk + 1) * Hh + h] };
      v2f a0 = { As[buf][mrow][kk],      As[buf][mrow][kk + 1] };
      v2f a1 = { As[buf][16 + mrow][kk], As[buf][16 + mrow][kk + 1] };
      acc0 = wmma_f32(a0, b, acc0);
      acc1 = wmma_f32(a1, b, acc1);
    }
    if (c + 1 < NC) {                 // land prefetch in the other buffer
#pragma unroll
      for (int j = 0; j < 4; ++j) *(float4*)&As[buf ^ 1][rr[j]][cc[j]] = rbuf[j];
    }
    __syncthreads();
  }

  const float bias  = bout[h];
  const int   mbase = (lane >> 4) * 8;
#pragma unroll
  for (int g = 0; g < 8; ++g) {
    Cs[mbase + g][h]      = acc0[g] + bias;
    Cs[16 + mbase + g][h] = acc1[g] + bias;
  }
  __syncthreads();

  if (tid < 32) {
    float m = 0.f;
    for (int cidx = 0; cidx < Hh; ++cidx) m += Cs[tid][cidx];
    m *= (1.0f / Hh);
    float v = 0.f;
    for (int cidx = 0; cidx < Hh; ++cidx) { const float x = Cs[tid][cidx] - m; v += x * x; }
    v *= (1.0f / Hh);
    Cs[tid][256] = m;
    Cs[tid][257] = rsqrtf(v + 1e-5f);
  }
  __syncthreads();

  for (int i = tid; i < 32 * 256; i += 512) {
    const int r = i >> 8, cidx = i & 255;
    if (m0 + r < Nn) {
      const float x = (Cs[r][cidx] - Cs[r][256]) * Cs[r][257] * g3[cidx] + b3[cidx];
      out[(size_t)(m0 + r) * Hh + cidx] = fmaxf(x, 0.0f);
    }
  }
}

// ---------------------------------------------------------------------------
extern "C" void kernel_launch(void* const* d_in, const int* in_sizes, int n_in,
                              void* d_out, int out_size, void* d_ws, size_t ws_size,
                              hipStream_t stream) {
  const float* pro  = (const float*)d_in[0];   // (1, N, H)
  const float* roi  = (const float*)d_in[1];   // (R, N, H)
  const float* Wdyn = (const float*)d_in[2];   // (H, 2HD)
  const float* bdyn = (const float*)d_in[3];   // (2HD)
  const float* Wout = (const float*)d_in[4];   // (H*R, H)
  const float* bout = (const float*)d_in[5];   // (H)
  const float* g1   = (const float*)d_in[6];
  const float* b1   = (const float*)d_in[7];
  const float* g2   = (const float*)d_in[8];
  const float* b2   = (const float*)d_in[9];
  const float* g3   = (const float*)d_in[10];
  const float* b3   = (const float*)d_in[11];
  float*       out  = (float*)d_out;           // (N, H)

  // workspace: params (2000 x 32768 f32 = 262 MB), f2 (2000 x 12544 f32 = 100 MB)
  float* params = (float*)d_ws;
  float* f2     = params + (size_t)Nn * PW;

  params_gemm_k<<<dim3(Nn / 80, PW / 128), 256, 0, stream>>>(pro, Wdyn, bdyn, params);
  dynconv_k<<<Nn, 256, 0, stream>>>(roi, params, g1, b1, g2, b2, f2);
  outlayer_k<<<(Nn + 31) / 32, 512, 0, stream>>>(f2, Wout, bout, g3, b3, out);
}